// SoftHistogramLoss_47820165874067
// MI455X (gfx1250) — compile-verified
//
#include <hip/hip_runtime.h>
#include <math.h>

// ---------------------------------------------------------------------------
// SoftHistogramLoss for MI455X (gfx1250, wave32).
//
// Roofline: ~335M 9-dim circular-distance evals, all data L2-resident
// (patches ~1MB, bins 72KB) -> VALU-bound, not bandwidth-bound.
// d^2 = diff^2 + min(0, 1-2|diff|); the bilinear diff^2 term is computed as
// Sp2 + Sb2 - 2*(A x B) with V_WMMA_F32_16X16X4_F32 (K=9 padded to 12 -> 3
// chained WMMAs per 16x16 (patch,bin) tile), co-executing with the VALU
// wrap-correction pass. Per-bin partials accumulate via global_atomic_add_f32.
// ---------------------------------------------------------------------------

typedef __attribute__((ext_vector_type(2))) float v2f;
typedef __attribute__((ext_vector_type(4))) float v4f;
typedef __attribute__((ext_vector_type(8))) float v8f;

#define HW     65536
#define NBINS  2048
#define SCALE  1300500.0f      /* (1/BIN_WIDTH^2)/TEMPERATURE = 255^2/0.05 */
#define EPSF   1.1920929e-07f
#define SENT   1.0e4f          /* sentinel for padded patches -> exp underflows to 0 */

// -------- kernel 1: gray conversion + zero accumulators ---------------------
__global__ void k_gray_init(const float* __restrict__ img, float* __restrict__ gray,
                            float* __restrict__ rawD, float* __restrict__ rawC) {
  int i = blockIdx.x * blockDim.x + threadIdx.x;
  if (i < HW) gray[i] = (img[i] + img[HW + i] + img[2 * HW + i]) * (1.0f / 3.0f);
  if (i < NBINS) { rawD[i] = 0.0f; rawC[i] = 0.0f; }
}

// -------- kernel 2: gather image patches into [n][16] WMMA-friendly rows ----
// row layout: floats 0..8 = patch dims, 9..11 = 0 (K-padding), 12 = sum(p^2)
__global__ void k_pack_cur(const float* __restrict__ gray, const int* __restrict__ pidx,
                           float* __restrict__ packed, int npat, int npad) {
  int n = blockIdx.x * blockDim.x + threadIdx.x;
  if (n >= npad) return;
  float* o = packed + (size_t)n * 16;
  float s2 = 0.0f;
  if (n < npat) {
#pragma unroll
    for (int d = 0; d < 9; ++d) {
      float v = gray[pidx[n * 9 + d]];
      o[d] = v; s2 = fmaf(v, v, s2);
    }
  } else {
#pragma unroll
    for (int d = 0; d < 9; ++d) o[d] = SENT;   // exp(-SCALE * ~9e8) == 0
    s2 = 9.0f * SENT * SENT;
  }
  o[9] = 0.0f; o[10] = 0.0f; o[11] = 0.0f;
  o[12] = s2;  o[13] = 0.0f; o[14] = 0.0f; o[15] = 0.0f;
}

// -------- kernel 3: transpose desired_patches [9][2048] -> [n][16] rows -----
__global__ void k_pack_des(const float* __restrict__ des, float* __restrict__ packed) {
  int n = blockIdx.x * blockDim.x + threadIdx.x;
  if (n >= NBINS) return;
  float* o = packed + (size_t)n * 16;
  float s2 = 0.0f;
#pragma unroll
  for (int d = 0; d < 9; ++d) {
    float v = des[d * NBINS + n];
    o[d] = v; s2 = fmaf(v, v, s2);
  }
  o[9] = 0.0f; o[10] = 0.0f; o[11] = 0.0f;
  o[12] = s2;  o[13] = 0.0f; o[14] = 0.0f; o[15] = 0.0f;
}

// -------- kernel 4: soft-KDE histogram, WMMA cross-term + VALU wrap term ----
// Wave gw owns bin tile (gw % 128) and patch-tile stride class (gw / 128).
// All 8 waves of a block share the same patch-tile sequence, so the block
// stages one 16x16-float patch tile into LDS per iteration.
__global__ void __launch_bounds__(256)
k_hist(const float* __restrict__ packed, const float* __restrict__ bins,
       float* __restrict__ raw, int nTiles, int nChunks) {
  __shared__ float tile[16 * 16];

  const int tid  = threadIdx.x;
  const int lane = tid & 31;
  const int wid  = tid >> 5;
  const int gw      = blockIdx.x * 8 + wid;
  const int binTile = gw & 127;
  const int chunk   = gw >> 7;              // uniform across the block
  const int col  = lane & 15;
  const bool hi  = lane >= 16;
  const int koff = hi ? 2 : 0;
  const int bcol = binTile * 16 + col;

  // lane-resident bin column: 9 dims (+3 zero pad), sum(b^2), B-fragments
  float bv[12];
#pragma unroll
  for (int d = 0; d < 9; ++d) bv[d] = bins[d * NBINS + bcol];
  bv[9] = 0.0f; bv[10] = 0.0f; bv[11] = 0.0f;
  float sumB2 = 0.0f;
#pragma unroll
  for (int d = 0; d < 9; ++d) sumB2 = fmaf(bv[d], bv[d], sumB2);

  // B fragment (4x16, f32): lane holds column N=col, K = koff+{0,1} per VGPR pair
  v2f bf0, bf1, bf2;
  bf0[0] = bv[0 + koff]; bf0[1] = bv[1 + koff];
  bf1[0] = bv[4 + koff]; bf1[1] = bv[5 + koff];
  bf2[0] = bv[8 + koff]; bf2[1] = bv[9 + koff];

  float hacc = 0.0f;

  for (int t = chunk; t < nTiles; t += nChunks) {
    __syncthreads();                         // protect tile reuse
    if (tid < 64) {                          // 16 rows x 16 floats = 64 float4
      const v4f* src = (const v4f*)(packed + (size_t)t * 256);
      ((v4f*)tile)[tid] = src[tid];
    }
    __syncthreads();

    // A fragment (16x4, f32): lane holds row M=col, K = koff+{0,1} per VGPR pair
    v2f a0, a1, a2;
    a0[0] = tile[col * 16 + 0 + koff]; a0[1] = tile[col * 16 + 1 + koff];
    a1[0] = tile[col * 16 + 4 + koff]; a1[1] = tile[col * 16 + 5 + koff];
    a2[0] = tile[col * 16 + 8 + koff]; a2[1] = tile[col * 16 + 9 + koff];

    // cross[m][n] = sum_k p[m][k] * b[k][n], K padded to 12 with zeros
    v8f c = {};
    c = __builtin_amdgcn_wmma_f32_16x16x4_f32(false, a0, false, bf0, (short)0, c, false, false);
    c = __builtin_amdgcn_wmma_f32_16x16x4_f32(false, a1, false, bf1, (short)0, c, false, false);
    c = __builtin_amdgcn_wmma_f32_16x16x4_f32(false, a2, false, bf2, (short)0, c, false, false);

    // wrap correction + exp over the 8 C rows this lane owns
#pragma unroll
    for (int v = 0; v < 8; ++v) {
      const int m2 = v + (hi ? 8 : 0);       // C/D layout: VGPR v -> row v / v+8
      const float* row = tile + m2 * 16;
      float corr = 0.0f;
#pragma unroll
      for (int d = 0; d < 9; ++d) {
        float df = row[d] - bv[d];
        float u  = fmaf(-2.0f, fabsf(df), 1.0f);   // 1 - 2|diff|
        corr += fminf(u, 0.0f);
      }
      float dist2 = row[12] + sumB2 - 2.0f * c[v] + corr;
      hacc += __expf(-SCALE * dist2);
    }
  }

  atomicAdd(raw + bcol, hacc);               // global_atomic_add_f32
}

// -------- kernel 5: normalizer + clip + extra bin + KL mean -----------------
__global__ void __launch_bounds__(256)
k_final(const float* __restrict__ rawD, const float* __restrict__ rawC,
        float* __restrict__ out, int npat) {
  __shared__ float red[256];
  const int tid = threadIdx.x;

  float s = 0.0f;
  for (int b = tid; b < NBINS; b += 256) s += rawD[b];
  red[tid] = s; __syncthreads();
  for (int o = 128; o > 0; o >>= 1) { if (tid < o) red[tid] += red[tid + o]; __syncthreads(); }
  const float S = red[0];
  __syncthreads();

  // desired_hist_b = rawD_b / S ; cur_hist_b = rawC_b * N_D^2 / (S * P^2)
  const float invS   = 1.0f / S;
  const float fp     = (float)npat;
  const float curFac = (2048.0f * 2048.0f) * invS / (fp * fp);

  float sumHd = 0.0f, sumHc = 0.0f, klsum = 0.0f;
  for (int b = tid; b < NBINS; b += 256) {
    float hd = fminf(fmaxf(rawD[b] * invS,   0.0f), 1.0f);
    float hc = fminf(fmaxf(rawC[b] * curFac, 0.0f), 1.0f);
    sumHd += hd; sumHc += hc;
    if (hd > 0.0f) klsum += hd * (logf(hd + EPSF) - logf(hc + EPSF));
  }
  red[tid] = sumHd; __syncthreads();
  for (int o = 128; o > 0; o >>= 1) { if (tid < o) red[tid] += red[tid + o]; __syncthreads(); }
  const float Hd = red[0]; __syncthreads();
  red[tid] = sumHc; __syncthreads();
  for (int o = 128; o > 0; o >>= 1) { if (tid < o) red[tid] += red[tid + o]; __syncthreads(); }
  const float Hc = red[0]; __syncthreads();
  red[tid] = klsum; __syncthreads();
  for (int o = 128; o > 0; o >>= 1) { if (tid < o) red[tid] += red[tid + o]; __syncthreads(); }

  if (tid == 0) {
    float kl  = red[0];
    float exd = 1.0f - fminf(1.0f, Hd);
    float exc = 1.0f - fminf(1.0f, Hc);
    if (exd > 0.0f) kl += exd * (logf(exd + EPSF) - logf(exc + EPSF));
    out[0] = kl / 2049.0f;
  }
}

// ---------------------------------------------------------------------------
extern "C" void kernel_launch(void* const* d_in, const int* in_sizes, int n_in,
                              void* d_out, int out_size, void* d_ws, size_t ws_size,
                              hipStream_t stream) {
  const float* img  = (const float*)d_in[0];   // [3*65536]
  const float* bins = (const float*)d_in[1];   // [9*2048]
  const float* des  = (const float*)d_in[2];   // [9*2048]
  const int*   pidx = (const int*)d_in[3];     // [npat*9]

  const int npat    = in_sizes[3] / 9;                 // 16129
  const int nTilesC = (npat + 15) / 16;                // 1009
  const int npadC   = nTilesC * 16;                    // 16144
  const int nTilesD = NBINS / 16;                      // 128

  float* w     = (float*)d_ws;
  float* gray  = w;                 w += HW;           // 65536
  float* rawD  = w;                 w += NBINS;
  float* rawC  = w;                 w += NBINS;
  float* packD = w;                 w += NBINS * 16;
  float* packC = w;                 /* npadC * 16 */

  k_gray_init<<<(HW + 255) / 256, 256, 0, stream>>>(img, gray, rawD, rawC);
  k_pack_des<<<(NBINS + 255) / 256, 256, 0, stream>>>(des, packD);
  k_pack_cur<<<(npadC + 255) / 256, 256, 0, stream>>>(gray, pidx, packC, npat, npadC);

  // hist grid: 16 * nChunks blocks of 8 waves; wave -> (binTile, chunk)
  const int chunksD = 8;   // 1024 waves for the 128x128-tile desired pass
  const int chunksC = 32;  // 4096 waves for the 128x1009-tile image pass
  k_hist<<<16 * chunksD, 256, 0, stream>>>(packD, bins, rawD, nTilesD, chunksD);
  k_hist<<<16 * chunksC, 256, 0, stream>>>(packC, bins, rawC, nTilesC, chunksC);

  k_final<<<1, 256, 0, stream>>>(rawD, rawC, (float*)d_out, npat);
}